// BatchedPiecewiseLinearHCPathOptimizer_63032940036321
// MI455X (gfx1250) — compile-verified
//
#include <hip/hip_runtime.h>
#include <math.h>

typedef __attribute__((ext_vector_type(2))) float v2f;
typedef __attribute__((ext_vector_type(8))) float v8f;

#define R_CNT 64
#define K_CNT 32
#define M_CNT 64
#define NS 15                 // Sylvester matrix size (2*8-1)
#define SYL_STRIDE 453        // odd float stride per-thread region (>= 15*15*2=450)
#define GAMMA_PITCH 17        // 16 cols + 1 pad
#define OFF_P0 0              // 16 floats
#define OFF_P1 16             // 16 floats
#define OFF_W 32              // 64 floats (w per sample m)
#define OFF_GAMMA 96          // 64*17 = 1088 floats
#define OFF_SYL (OFF_GAMMA + M_CNT * GAMMA_PITCH)        // 1184
#define SMEM_FLOATS (OFF_SYL + 32 * SYL_STRIDE)          // 1184 + 14496 = 15680 (62.7 KB)

#define LOG_EPS12   (-27.631021115928547f)  // log(1e-12) == 2*log(1e-6)
#define LOG_EPS_SOFT (-13.815510557964274f) // log(1e-6)

__device__ __forceinline__ float loadP(const float* __restrict__ P0,
                                       const float* __restrict__ PK,
                                       const float* __restrict__ Pm,
                                       int r, int i, int j) {
  // P[r, i, :, :] flattened over (DEGREE,2) -> j in [0,16)
  if (i == 0)  return P0[r * 16 + j];
  if (i == 32) return PK[r * 16 + j];
  return Pm[(r * 31 + (i - 1)) * 16 + j];
}

__global__ void hc_seg_kernel(const float* __restrict__ P0,
                              const float* __restrict__ PK,
                              const float* __restrict__ Pm,
                              float* __restrict__ partial) {
  extern __shared__ float sm[];
  const int b = blockIdx.x;        // 0..2047
  const int r = b >> 5;
  const int k = b & 31;
  const int tid = threadIdx.x;     // 0..31 (one wave)
  const int lane = tid;

  // ---- load segment endpoints (flattened 16 floats each) ----
  if (tid < 16) {
    sm[OFF_P0 + tid] = loadP(P0, PK, Pm, r, k, tid);
    sm[OFF_P1 + tid] = loadP(P0, PK, Pm, r, k + 1, tid);
  }
  __syncthreads();

  // ---- WMMA interpolation: gamma(m, j) = (1-t_m)*P0f[j] + t_m*P1f[j] ----
  // B (4x16 f32): row0 = P0flat, row1 = P1flat, rows 2,3 = 0.
  //   VGPR0 holds K=0 (lanes 0-15) / K=2 (lanes 16-31); VGPR1 holds K=1 / K=3.
  v2f bop;
  bop.x = (lane < 16) ? sm[OFF_P0 + (lane & 15)] : 0.0f;
  bop.y = (lane < 16) ? sm[OFF_P1 + (lane & 15)] : 0.0f;

#pragma unroll
  for (int tile = 0; tile < 4; ++tile) {
    const int mbase = tile * 16;
    // A (16x4 f32): lanes 0-15 carry M=lane, VGPR0=K0=(1-t), VGPR1=K1=t;
    //               lanes 16-31 carry K=2,3 -> zero padded.
    const float t = (float)(mbase + (lane & 15)) * (1.0f / 63.0f);
    v2f aop;
    aop.x = (lane < 16) ? (1.0f - t) : 0.0f;
    aop.y = (lane < 16) ? t : 0.0f;
    v8f c = {};
    c = __builtin_amdgcn_wmma_f32_16x16x4_f32(false, aop, false, bop,
                                              (short)0, c, false, false);
    // C tile layout: element (M = i + 8*(lane>=16), N = lane&15) in c[i]
    const int mrow = mbase + ((lane >= 16) ? 8 : 0);
    const int ncol = lane & 15;
#pragma unroll
    for (int i = 0; i < 8; ++i)
      sm[OFF_GAMMA + (mrow + i) * GAMMA_PITCH + ncol] = c[i];
  }
  __syncthreads();

  // ---- per-thread: two samples (m = tid, tid+32), each a 15x15 complex LU ----
  float* S = &sm[OFF_SYL + tid * SYL_STRIDE]; // (re,im) pairs, row-major

  for (int pass = 0; pass < 2; ++pass) {
    const int m = tid + pass * 32;

    // monic coefficients (descending): [1, c1..c8], c_d = gamma[2d-2] + i*gamma[2d-1]
    float cr[9], ci[9];
    cr[0] = 1.0f; ci[0] = 0.0f;
#pragma unroll
    for (int d = 0; d < 8; ++d) {
      cr[1 + d] = sm[OFF_GAMMA + m * GAMMA_PITCH + 2 * d];
      ci[1 + d] = sm[OFF_GAMMA + m * GAMMA_PITCH + 2 * d + 1];
    }
    float der[8], dei[8];
#pragma unroll
    for (int j = 0; j < 8; ++j) {
      const float wgt = (float)(8 - j);
      der[j] = cr[j] * wgt; dei[j] = ci[j] * wgt;
    }

    // zero + fill Sylvester(f, f') : rows 0..6 = coeffs, rows 7..14 = dcoef
    for (int i = 0; i < NS * NS * 2; ++i) S[i] = 0.0f;
#pragma unroll
    for (int i = 0; i < 7; ++i)
#pragma unroll
      for (int j = 0; j <= 8; ++j) {
        S[(i * NS + i + j) * 2]     = cr[j];
        S[(i * NS + i + j) * 2 + 1] = ci[j];
      }
#pragma unroll
    for (int i = 0; i < 8; ++i)
#pragma unroll
      for (int j = 0; j < 8; ++j) {
        S[((7 + i) * NS + i + j) * 2]     = der[j];
        S[((7 + i) * NS + i + j) * 2 + 1] = dei[j];
      }

    // complex Gaussian elimination with partial pivoting; log|det| = sum log|pivot|
    float logdet = 0.0f;
    for (int p = 0; p < NS; ++p) {
      int piv = p;
      float best = S[(p * NS + p) * 2] * S[(p * NS + p) * 2] +
                   S[(p * NS + p) * 2 + 1] * S[(p * NS + p) * 2 + 1];
      for (int i = p + 1; i < NS; ++i) {
        const float xr = S[(i * NS + p) * 2], xi = S[(i * NS + p) * 2 + 1];
        const float m2 = xr * xr + xi * xi;
        if (m2 > best) { best = m2; piv = i; }
      }
      if (piv != p) {
        for (int j = p; j < NS; ++j) {
          const float tr = S[(p * NS + j) * 2], ti = S[(p * NS + j) * 2 + 1];
          S[(p * NS + j) * 2]       = S[(piv * NS + j) * 2];
          S[(p * NS + j) * 2 + 1]   = S[(piv * NS + j) * 2 + 1];
          S[(piv * NS + j) * 2]     = tr;
          S[(piv * NS + j) * 2 + 1] = ti;
        }
      }
      const float dre = S[(p * NS + p) * 2], dim = S[(p * NS + p) * 2 + 1];
      const float m2 = dre * dre + dim * dim;
      logdet += 0.5f * logf(m2);
      const float inv = 1.0f / m2;
      for (int i = p + 1; i < NS; ++i) {
        const float are = S[(i * NS + p) * 2], aim = S[(i * NS + p) * 2 + 1];
        const float fr = (are * dre + aim * dim) * inv;
        const float fi = (aim * dre - are * dim) * inv;
        for (int j = p + 1; j < NS; ++j) {
          const float sre = S[(p * NS + j) * 2], sim = S[(p * NS + j) * 2 + 1];
          S[(i * NS + j) * 2]     -= fr * sre - fi * sim;
          S[(i * NS + j) * 2 + 1] -= fr * sim + fi * sre;
        }
      }
    }
    if (!isfinite(logdet)) logdet = LOG_EPS12;
    logdet = fmaxf(logdet, LOG_EPS12);

    // soft-abs weight chain (all fp32, matching reference)
    const float a2 = 2.0f * logdet, b2 = LOG_EPS12; // 2*log(delta) = log(1e-12)
    const float mx2 = fmaxf(a2, b2), mn2 = fminf(a2, b2);
    const float log_soft = 0.5f * (mx2 + log1pf(expf(mn2 - mx2)));
    const float mx3 = fmaxf(log_soft, LOG_EPS_SOFT), mn3 = fminf(log_soft, LOG_EPS_SOFT);
    const float lse = mx3 + log1pf(expf(mn3 - mx3));
    sm[OFF_W + m] = expf(-lse * 0.125f);
  }
  __syncthreads();

  if (tid == 0) {
    float s = 0.0f;
    for (int i = 0; i < M_CNT; ++i) s += sm[OFF_W + i]; // fixed order -> deterministic
    const float wmean = s * (1.0f / (float)M_CNT);
    float sl2 = 0.0f;
    for (int j = 0; j < 16; ++j) {
      const float d = sm[OFF_P1 + j] - sm[OFF_P0 + j];
      sl2 += d * d;
    }
    partial[b] = sqrtf(sl2) * wmean;
  }
}

__global__ void hc_final_kernel(const float* __restrict__ P0,
                                const float* __restrict__ PK,
                                const float* __restrict__ Pm,
                                const float* __restrict__ partial,
                                float* __restrict__ out) {
  __shared__ float red[256];
  const int tid = threadIdx.x;

  float acc = 0.0f;
  for (int i = tid; i < R_CNT * K_CNT; i += 256) acc += partial[i];

  // smoothness: sum over r, i in [0,31), j in [0,16) of (P[i+2]-2P[i+1]+P[i])^2
  float smv = 0.0f;
  for (int idx = tid; idx < R_CNT * 31 * 16; idx += 256) {
    const int j = idx & 15;
    const int t2 = idx >> 4;
    const int i = t2 % 31;
    const int r = t2 / 31;
    const float a = loadP(P0, PK, Pm, r, i, j);
    const float m = loadP(P0, PK, Pm, r, i + 1, j);
    const float c = loadP(P0, PK, Pm, r, i + 2, j);
    const float d2 = c - 2.0f * m + a;
    smv += d2 * d2;
  }
  red[tid] = acc + 0.1f * smv;
  __syncthreads();
  for (int s = 128; s > 0; s >>= 1) {
    if (tid < s) red[tid] += red[tid + s];
    __syncthreads();
  }
  if (tid == 0) out[0] = red[0];
}

extern "C" void kernel_launch(void* const* d_in, const int* in_sizes, int n_in,
                              void* d_out, int out_size, void* d_ws, size_t ws_size,
                              hipStream_t stream) {
  (void)in_sizes; (void)n_in; (void)out_size; (void)ws_size;
  const float* P0 = (const float*)d_in[0];
  const float* PK = (const float*)d_in[1];
  const float* Pm = (const float*)d_in[2];
  float* partial = (float*)d_ws;          // 2048 floats
  float* out = (float*)d_out;

  const size_t smem = (size_t)SMEM_FLOATS * sizeof(float);
  hc_seg_kernel<<<R_CNT * K_CNT, 32, smem, stream>>>(P0, PK, Pm, partial);
  hc_final_kernel<<<1, 256, 0, stream>>>(P0, PK, Pm, partial, out);
}